// LightGCNConv_28089086116173
// MI455X (gfx1250) — compile-verified
//
#include <hip/hip_runtime.h>

// LightGCN propagation: out = D^{-1/2} A D^{-1/2} x, COO edges.
// N=100000 nodes, E=1600000 edges, D=64 fp32 features.
//
// Roofline: 0.2 GFLOP vs ~0.8GB of L2-side traffic (x is 25.6MB -> resident in
// the 192MB L2) => bandwidth bound; WMMA gives ~6% useful occupancy on this
// pattern and does not touch the gather-bandwidth limiter, so it is not used.
// The naive scatter would need E*D = 102.4M fp32 atomics (atomic-throughput
// bound); instead we counting-sort edges into CSR (3.2M int atomics total) and
// do atomic-free per-row register accumulation. Edge streams are staged into
// LDS with gfx1250 async-to-LDS loads (ASYNCcnt), double buffered per wave.

#define D_FEAT 64
#define CPW 128              // edges per wave-chunk in the perm-build kernel
#define WAVES_PER_BLOCK 8    // 256 threads = 8 wave32
#define BUILD_BLOCKS 256
#define SCAN_BLK 1024        // elements per scan block (256 threads x 4)

// ---- gfx1250 async-to-LDS helpers ------------------------------------------
// Flat LDS addresses on CDNA5 carry the LDS byte offset in the low 32 bits
// (aperture in the high 32), so truncating a generic shared pointer yields the
// VDST offset the async instruction wants.
__device__ __forceinline__ void async_ld_b128(const void* gptr, void* ldsptr) {
  unsigned ldso = (unsigned)(unsigned long long)ldsptr;
  unsigned long long ga = (unsigned long long)gptr;
  asm volatile("global_load_async_to_lds_b128 %0, %1, off"
               :
               : "v"(ldso), "v"(ga)
               : "memory");
}
#define S_WAIT_ASYNCCNT_0() asm volatile("s_wait_asynccnt 0x0" ::: "memory")
#define S_WAIT_ASYNCCNT_2() asm volatile("s_wait_asynccnt 0x2" ::: "memory")
#define S_WAIT_DSCNT_0()    asm volatile("s_wait_dscnt 0x0" ::: "memory")

// ---- utility kernels --------------------------------------------------------
__global__ __launch_bounds__(256) void zero_u32(unsigned* __restrict__ p, int n) {
  int i = blockIdx.x * blockDim.x + threadIdx.x;
  int s = gridDim.x * blockDim.x;
  for (; i < n; i += s) p[i] = 0u;
}

__global__ __launch_bounds__(256) void degree_kernel(const int* __restrict__ row,
                                                     unsigned* __restrict__ deg, int E) {
  int i = blockIdx.x * blockDim.x + threadIdx.x;
  int s = gridDim.x * blockDim.x;
  for (; i < E; i += s) atomicAdd(&deg[row[i]], 1u);
}

__global__ __launch_bounds__(256) void dinv_kernel(const unsigned* __restrict__ deg,
                                                   float* __restrict__ dinv, int N) {
  int i = blockIdx.x * blockDim.x + threadIdx.x;
  if (i < N) {
    unsigned d = deg[i];
    // reference: where(deg>0, rsqrt(max(deg,1)), 0); deg>0 => max is identity
    dinv[i] = (d > 0u) ? (1.0f / sqrtf((float)d)) : 0.0f;
  }
}

// ---- hierarchical exclusive scan (3 tiny kernels, fully parallel) -----------
// scan1: each block scans SCAN_BLK elements (4 per thread sequential + one
//        256-wide LDS Hillis-Steele), writes block-local exclusive prefixes
//        into off[] and the block total into blkSum[b].
__global__ __launch_bounds__(256) void scan1_kernel(const unsigned* __restrict__ deg,
                                                    unsigned* __restrict__ off,
                                                    unsigned* __restrict__ blkSum, int N) {
  __shared__ unsigned s[256];
  const int tid = threadIdx.x;
  const int base = blockIdx.x * SCAN_BLK + tid * 4;
  unsigned v[4];
#pragma unroll
  for (int k = 0; k < 4; ++k) v[k] = (base + k < N) ? deg[base + k] : 0u;
  const unsigned tsum = v[0] + v[1] + v[2] + v[3];
  s[tid] = tsum;
  __syncthreads();
  for (int o = 1; o < 256; o <<= 1) {
    unsigned t = (tid >= o) ? s[tid - o] : 0u;
    __syncthreads();
    s[tid] += t;
    __syncthreads();
  }
  unsigned p = s[tid] - tsum;  // exclusive prefix of this thread's group of 4
  if (tid == 255) blkSum[blockIdx.x] = s[255];
#pragma unroll
  for (int k = 0; k < 4; ++k) {
    if (base + k < N) off[base + k] = p;
    p += v[k];
  }
}

// scan2: one block scans the (~98) block sums exclusively, writes off[N]=total.
__global__ __launch_bounds__(256) void scan2_kernel(unsigned* __restrict__ blkSum,
                                                    unsigned* __restrict__ blkOff,
                                                    unsigned* __restrict__ off,
                                                    int nB, int N) {
  __shared__ unsigned s[256];
  const int tid = threadIdx.x;
  if (nB <= 256) {
    const unsigned vv = (tid < nB) ? blkSum[tid] : 0u;
    s[tid] = vv;
    __syncthreads();
    for (int o = 1; o < 256; o <<= 1) {
      unsigned t = (tid >= o) ? s[tid - o] : 0u;
      __syncthreads();
      s[tid] += t;
      __syncthreads();
    }
    if (tid < nB) blkOff[tid] = s[tid] - vv;
    if (tid == 255) off[N] = s[255];  // == E
  } else if (tid == 0) {              // generic fallback, still tiny
    unsigned acc = 0u;
    for (int b = 0; b < nB; ++b) {
      blkOff[b] = acc;
      acc += blkSum[b];
    }
    off[N] = acc;
  }
}

// scan3: add block offsets in place; duplicate into write cursors.
__global__ __launch_bounds__(256) void scan3_kernel(unsigned* __restrict__ off,
                                                    const unsigned* __restrict__ blkOff,
                                                    unsigned* __restrict__ cur, int N) {
  int i = blockIdx.x * blockDim.x + threadIdx.x;
  int s = gridDim.x * blockDim.x;
  for (; i < N; i += s) {
    const unsigned o = off[i] + blkOff[i / SCAN_BLK];
    off[i] = o;
    cur[i] = o;
  }
}

// ---- permutation build (counting-sort scatter) ------------------------------
// One wave owns a 128-edge chunk. row/col streams are staged into the wave's
// private LDS slice with async b128 loads, double buffered (2 loads per chunk,
// in-order completion => s_wait_asynccnt 2 retires the current chunk). Each
// lane then consumes exactly the 4 edges it prefetched: pos = cursor fetch-add
// (one int atomic per edge), and meta[pos] = {col, norm} as an 8B record.
__global__ __launch_bounds__(256) void build_perm_kernel(const int* __restrict__ row,
                                                         const int* __restrict__ col,
                                                         const float* __restrict__ dinv,
                                                         unsigned* __restrict__ cur,
                                                         int2* __restrict__ meta, int E) {
  __shared__ int sRow[2][WAVES_PER_BLOCK][CPW];
  __shared__ int sCol[2][WAVES_PER_BLOCK][CPW];

  const int lane = threadIdx.x & 31;
  const int w    = threadIdx.x >> 5;
  const int gw   = (int)((blockIdx.x * blockDim.x + threadIdx.x) >> 5);
  const int nW   = (int)((gridDim.x * blockDim.x) >> 5);
  const int nChunks = E / CPW;

  auto issue = [&](int chunk, int b) {
    const int base = chunk * CPW + lane * 4;  // 4 int32 per lane (b128)
    async_ld_b128(row + base, &sRow[b][w][lane * 4]);
    async_ld_b128(col + base, &sCol[b][w][lane * 4]);
  };
  auto process_edge = [&](int r, int c) {
    const unsigned pos = atomicAdd(&cur[r], 1u);
    const float nm = dinv[r] * dinv[c];
    meta[pos] = make_int2(c, __float_as_int(nm));
  };

  int buf = 0;
  if (gw < nChunks) issue(gw, 0);  // wave-uniform branch: EXEC stays full
  for (int c = gw; c < nChunks; c += nW) {
    const int cnext = c + nW;
    if (cnext < nChunks) {
      S_WAIT_DSCNT_0();            // prior chunk's LDS reads retired (WAR)
      issue(cnext, buf ^ 1);
      S_WAIT_ASYNCCNT_2();         // current chunk's 2 loads done (in-order)
    } else {
      S_WAIT_ASYNCCNT_0();
    }
    const int4 r4 = *(const int4*)&sRow[buf][w][lane * 4];  // ds_load_b128
    const int4 c4 = *(const int4*)&sCol[buf][w][lane * 4];
    process_edge(r4.x, c4.x);
    process_edge(r4.y, c4.y);
    process_edge(r4.z, c4.z);
    process_edge(r4.w, c4.w);
    buf ^= 1;
  }

  // Tail edges (E % 128): direct global reads, one edge per thread.
  for (int t = nChunks * CPW + blockIdx.x * blockDim.x + threadIdx.x; t < E;
       t += gridDim.x * blockDim.x)
    process_edge(row[t], col[t]);
}

// ---- atomic-free per-row accumulation ---------------------------------------
// One wave per row. Lane l owns features {2l, 2l+1}; accumulate in registers
// over the row's contiguous CSR segment, single float2 store. Edge metadata
// addresses are wave-uniform (readfirstlane) -> scalar loads; x gathers hit
// the L2-resident feature matrix.
__global__ __launch_bounds__(256) void row_kernel(const unsigned* __restrict__ off,
                                                  const int2* __restrict__ meta,
                                                  const float* __restrict__ x,
                                                  float* __restrict__ out, int N) {
  const int lane = threadIdx.x & 31;
  const int w    = threadIdx.x >> 5;
  int r = blockIdx.x * WAVES_PER_BLOCK + w;
  if (r >= N) return;
  r = __builtin_amdgcn_readfirstlane(r);

  const unsigned j0 = off[r];
  const unsigned j1 = off[r + 1];
  const size_t fo = (size_t)(lane << 1);
  float2 acc = make_float2(0.0f, 0.0f);
#pragma unroll 4
  for (unsigned j = j0; j < j1; ++j) {
    const int2 m = meta[j];                 // wave-uniform 8B load
    const float nm = __int_as_float(m.y);
    const float2 xv = *(const float2*)(x + ((size_t)m.x << 6) + fo);
    acc.x += nm * xv.x;
    acc.y += nm * xv.y;
  }
  *(float2*)(out + ((size_t)r << 6) + fo) = acc;  // rows with deg==0 write 0
}

// ---- launcher ---------------------------------------------------------------
extern "C" void kernel_launch(void* const* d_in, const int* in_sizes, int n_in,
                              void* d_out, int out_size, void* d_ws, size_t ws_size,
                              hipStream_t stream) {
  const float* x  = (const float*)d_in[0];
  const int*   ei = (const int*)d_in[1];
  const int N = in_sizes[0] / D_FEAT;
  const int E = in_sizes[1] / 2;
  const int* row = ei;
  const int* col = ei + E;
  float* out = (float*)d_out;
  const int nB = (N + SCAN_BLK - 1) / SCAN_BLK;  // scan blocks (~98)

  // Workspace layout (bytes, all 4B-aligned; meta rounded to 8B):
  //   deg u32[N] | dinv f32[N] | off u32[N+1] | cur u32[N] |
  //   blkSum u32[nB] | blkOff u32[nB] | meta int2[E]        => ~14.4 MB
  char* ws = (char*)d_ws;
  size_t o = 0;
  unsigned* deg    = (unsigned*)(ws + o); o += (size_t)N * 4;
  float*    dinv   = (float*)(ws + o);    o += (size_t)N * 4;
  unsigned* off    = (unsigned*)(ws + o); o += (size_t)(N + 1) * 4;
  unsigned* cur    = (unsigned*)(ws + o); o += (size_t)N * 4;
  unsigned* blkSum = (unsigned*)(ws + o); o += (size_t)nB * 4;
  unsigned* blkOff = (unsigned*)(ws + o); o += (size_t)nB * 4;
  o = (o + 7) & ~(size_t)7;
  int2*     meta   = (int2*)(ws + o);

  // 1) zero degree counters
  zero_u32<<<128, 256, 0, stream>>>(deg, N);
  // 2) degree histogram (1.6M int atomics)
  degree_kernel<<<1024, 256, 0, stream>>>(row, deg, E);
  // 3) deg_inv_sqrt, matching the reference's where/max semantics
  dinv_kernel<<<(N + 255) / 256, 256, 0, stream>>>(deg, dinv, N);
  // 4) hierarchical exclusive scan -> CSR row offsets + write cursors
  scan1_kernel<<<nB, 256, 0, stream>>>(deg, off, blkSum, N);
  scan2_kernel<<<1, 256, 0, stream>>>(blkSum, blkOff, off, nB, N);
  scan3_kernel<<<(N + 255) / 256, 256, 0, stream>>>(off, blkOff, cur, N);
  // 5) counting-sort edges into CSR order with async LDS edge staging
  build_perm_kernel<<<BUILD_BLOCKS, 256, 0, stream>>>(row, col, dinv, cur, meta, E);
  // 6) atomic-free per-row gather/scale/accumulate
  row_kernel<<<(N + WAVES_PER_BLOCK - 1) / WAVES_PER_BLOCK, 256, 0, stream>>>(
      off, meta, x, out, N);
}